// ReverseDeformLayer_63075889709150
// MI455X (gfx1250) — compile-verified
//
#include <hip/hip_runtime.h>
#include <math.h>

typedef __attribute__((ext_vector_type(2))) float v2f;
typedef __attribute__((ext_vector_type(8))) float v8f;

// ---------------------------------------------------------------------------
// Kernel 1: one wave per (target-tile, source-chunk).
// d2[m][n] = ||s_n||^2 - 2 t_m . s_n computed as a single K=4 fp32 WMMA:
//   A (16x4): rows m -> [-2tx, -2ty, -2tz, 1.0]
//   B (4x16): cols n -> [ sx ,  sy ,  sz , ||s_n||^2]
//   C = 0 (loop-invariant zero registers; no per-tile broadcast movs)
// A layout (32-bit A 16x4): lanes 0-15 hold K=0,1 in VGPR0,1; lanes 16-31 K=2,3.
// B mirrored. C/D: VGPR r, lanes 0-15 -> row r, lanes 16-31 -> row 8+r,
// column = lane & 15.
// ---------------------------------------------------------------------------
__global__ void nn_tile_kernel(const float* __restrict__ src,
                               const float* __restrict__ tar,
                               float2* __restrict__ pairs,
                               int S, int nSrc, int nTiles) {
    const int lane  = threadIdx.x & 31;
    const int waveG = (int)((blockIdx.x * blockDim.x + threadIdx.x) >> 5);
    const int tileM = waveG / S;      // wave-uniform
    const int chunk = waveG % S;      // wave-uniform
    if (tileM >= nTiles) return;      // uniform exit: EXEC stays all-1s for WMMA

    const int  col = lane & 15;
    const bool hi  = lane >= 16;

    // A = [-2*t | 1] for this wave's 16 target rows.
    const int m = tileM * 16 + col;
    const float tx = tar[3 * m + 0];
    const float ty = tar[3 * m + 1];
    const float tz = tar[3 * m + 2];
    v2f a;
    a.x = hi ? (-2.0f * tz) : (-2.0f * tx);
    a.y = hi ? 1.0f         : (-2.0f * ty);

    v8f czero;
#pragma unroll
    for (int r = 0; r < 8; ++r) czero[r] = 0.0f;   // loop-invariant

    float best[8];
    int   bidx[8];
#pragma unroll
    for (int r = 0; r < 8; ++r) { best[r] = INFINITY; bidx[r] = 0; }

    const int tilesPerChunk = (nSrc >> 4) / S;
    const int tile0 = chunk * tilesPerChunk;

#pragma unroll 2
    for (int t = 0; t < tilesPerChunk; ++t) {
        const int n = ((tile0 + t) << 4) + col;     // this lane's source column
        const float sx = src[3 * n + 0];
        const float sy = src[3 * n + 1];
        const float sz = src[3 * n + 2];
        const float snorm = sx * sx + sy * sy + sz * sz;

        // B = [s | ||s||^2] for 16 source columns.
        v2f b;
        b.x = hi ? sz    : sx;
        b.y = hi ? snorm : sy;

        // d2 tile = A*B + 0  (fp32 WMMA, K=4)
        v8f d = __builtin_amdgcn_wmma_f32_16x16x4_f32(
            /*neg_a=*/false, a, /*neg_b=*/false, b,
            /*c_mod=*/(short)0, czero, /*reuse_a=*/false, /*reuse_b=*/false);

#pragma unroll
        for (int r = 0; r < 8; ++r) {
            const bool lt = d[r] < best[r];
            best[r] = lt ? d[r] : best[r];
            bidx[r] = lt ? n    : bidx[r];
        }
    }

    // Reduce the 16 columns within each half-wave (rows stay put: xor<16).
#pragma unroll
    for (int off = 8; off >= 1; off >>= 1) {
#pragma unroll
        for (int r = 0; r < 8; ++r) {
            const float ob = __shfl_xor(best[r], off, 32);
            const int   oi = __shfl_xor(bidx[r], off, 32);
            if (ob < best[r] || (ob == best[r] && oi < bidx[r])) {
                best[r] = ob;
                bidx[r] = oi;
            }
        }
    }

    // Lane 0 owns rows 0-7, lane 16 owns rows 8-15 of this tile.
    if (col == 0) {
        const int rowBase = tileM * 16 + (hi ? 8 : 0);
#pragma unroll
        for (int r = 0; r < 8; ++r) {
            const int mm = rowBase + r;
            pairs[(size_t)mm * S + chunk] = make_float2(best[r], __int_as_float(bidx[r]));
        }
    }
}

// ---------------------------------------------------------------------------
// Kernel 2: per target, combine the S chunk candidates (fixed order ->
// deterministic), gather the nearest source, emit 0.5*||s - t||^2.
// ---------------------------------------------------------------------------
__global__ void pick_and_diff_kernel(const float* __restrict__ src,
                                     const float* __restrict__ tar,
                                     const float2* __restrict__ pairs,
                                     float* __restrict__ sums,
                                     int S, int nTar) {
    const int m = blockIdx.x * blockDim.x + threadIdx.x;
    if (m >= nTar) return;

    float best = INFINITY;
    int   bi   = 0;
    for (int c = 0; c < S; ++c) {
        const float2 p = pairs[(size_t)m * S + c];
        const int i = __float_as_int(p.y);
        if (p.x < best || (p.x == best && i < bi)) { best = p.x; bi = i; }
    }

    const float dx = src[3 * bi + 0] - tar[3 * m + 0];
    const float dy = src[3 * bi + 1] - tar[3 * m + 1];
    const float dz = src[3 * bi + 2] - tar[3 * m + 2];
    sums[m] = 0.5f * (dx * dx + dy * dy + dz * dz);
}

// ---------------------------------------------------------------------------
// Kernel 3: deterministic single-block tree reduction of 16384 partials.
// ---------------------------------------------------------------------------
__global__ void reduce_kernel(const float* __restrict__ sums,
                              float* __restrict__ out, int n) {
    __shared__ float sm[256];
    float acc = 0.0f;
    for (int i = threadIdx.x; i < n; i += 256) acc += sums[i];
    sm[threadIdx.x] = acc;
    __syncthreads();
#pragma unroll
    for (int s = 128; s > 0; s >>= 1) {
        if ((int)threadIdx.x < s) sm[threadIdx.x] += sm[threadIdx.x + s];
        __syncthreads();
    }
    if (threadIdx.x == 0) out[0] = sm[0];
}

extern "C" void kernel_launch(void* const* d_in, const int* in_sizes, int n_in,
                              void* d_out, int out_size, void* d_ws, size_t ws_size,
                              hipStream_t stream) {
    const float* src = (const float*)d_in[0];   // src_V [16384,3] f32
    const float* tar = (const float*)d_in[1];   // tar_V [16384,3] f32
    float* out = (float*)d_out;                 // scalar f32

    const int nSrc = in_sizes[0] / 3;
    const int nTar = in_sizes[1] / 3;
    const int nTiles = nTar / 16;

    // Largest source-split whose scratch fits: more waves -> better latency
    // hiding for the load->WMMA->compare chain (target ~2 waves/SIMD).
    int S = 8;
    while (S > 1 &&
           ws_size < (size_t)nTar * S * sizeof(float2) + (size_t)nTar * sizeof(float))
        S >>= 1;

    float2* pairs = (float2*)d_ws;
    float*  sums  = (float*)((char*)d_ws + (size_t)nTar * S * sizeof(float2));

    const int wavesTotal = nTiles * S;
    const int block = 256;                               // 8 wave32 waves
    const int grid  = (wavesTotal * 32 + block - 1) / block;

    nn_tile_kernel<<<grid, block, 0, stream>>>(src, tar, pairs, S, nSrc, nTiles);
    pick_and_diff_kernel<<<(nTar + 255) / 256, 256, 0, stream>>>(src, tar, pairs, sums, S, nTar);
    reduce_kernel<<<1, 256, 0, stream>>>(sums, out, nTar);
}